// SPADELight_64467459113021
// MI455X (gfx1250) — compile-verified
//
#include <hip/hip_runtime.h>
#include <hip/hip_bf16.h>
#include <stdint.h>

// Problem constants (from reference setup_inputs)
constexpr int NN = 8;
constexpr int CC = 128;
constexpr int HH = 256;
constexpr int WW = 256;
constexpr int LL = 20;              // label_nc
constexpr int HWSZ = HH * WW;       // 65536 elements per (n,c) plane
constexpr float EPSV = 1e-5f;

typedef float vf4 __attribute__((ext_vector_type(4)));

// ---- CDNA5 async-to-LDS helpers -------------------------------------------
#if __has_builtin(__builtin_amdgcn_s_wait_asynccnt)
#define WAIT_ASYNC(n) __builtin_amdgcn_s_wait_asynccnt(n)
#else
#define WAIT_ASYNC(n) asm volatile("s_wait_asynccnt %0" :: "i"(n) : "memory")
#endif

// global_load_async_to_lds_b128: VDST = per-lane LDS byte address (VGPR),
// VADDR = per-lane 32-bit byte offset (VGPR), SADDR = uniform 64-bit base.
#define ASYNC_LOAD_B128(ldsaddr, voff, sbase)                                  \
  asm volatile("global_load_async_to_lds_b128 %0, %1, %2"                      \
               :: "v"(ldsaddr), "v"(voff), "s"(sbase) : "memory")

// ---------------------------------------------------------------------------
// Kernel 1: per-pixel argmax over L segmap channels -> uint8 class ids.
// One thread handles 4 consecutive pixels (float4 loads, coalesced).
// segmap is read exactly once -> nontemporal loads keep it out of L2 so the
// x planes (needed twice by kernel 2) stay resident.
// ---------------------------------------------------------------------------
__global__ __launch_bounds__(256)
void spade_argmax_kernel(const float* __restrict__ seg,
                         unsigned char* __restrict__ cls) {
  const int g = blockIdx.x * blockDim.x + threadIdx.x;   // float4-group index
  const int total = NN * (HWSZ / 4);
  if (g >= total) return;

  const int n  = g / (HWSZ / 4);
  const int s4 = g % (HWSZ / 4);
  const vf4* seg4 = reinterpret_cast<const vf4*>(seg);
  const size_t base = (size_t)n * LL * (HWSZ / 4) + s4;

  vf4 best = __builtin_nontemporal_load(&seg4[base]);    // channel 0
  int b0 = 0, b1 = 0, b2 = 0, b3 = 0;
#pragma unroll
  for (int l = 1; l < LL; ++l) {
    const vf4 v = __builtin_nontemporal_load(&seg4[base + (size_t)l * (HWSZ / 4)]);
    if (v.x > best.x) { best.x = v.x; b0 = l; }
    if (v.y > best.y) { best.y = v.y; b1 = l; }
    if (v.z > best.z) { best.z = v.z; b2 = l; }
    if (v.w > best.w) { best.w = v.w; b3 = l; }
  }
  const unsigned packed = (unsigned)b0 | ((unsigned)b1 << 8) |
                          ((unsigned)b2 << 16) | ((unsigned)b3 << 24);
  reinterpret_cast<unsigned*>(cls)[g] = packed;
}

// ---------------------------------------------------------------------------
// Kernel 2: one block (256 threads = 8 wave32) per (n,c) plane.
// Pass 1: stream plane via per-wave 4-deep async-to-LDS ring, accumulate
//         sum / sum-of-squares (ASYNCcnt-paced, wave-private buffers, no
//         barriers). Loads are RT so the plane lands in L2.
// Pass 2: re-read plane (L2 resident), fuse normalize + class affine,
//         nontemporal stores so the write stream doesn't evict L2.
// ---------------------------------------------------------------------------
__global__ __launch_bounds__(256)
void spade_norm_kernel(const float* __restrict__ x,
                       const unsigned char* __restrict__ cls,
                       const float* __restrict__ weight,
                       const float* __restrict__ bias,
                       float* __restrict__ out) {
  const int bx  = blockIdx.x;          // n*CC + c
  const int n   = bx / CC;
  const int c   = bx % CC;
  const int tid = threadIdx.x;         // 0..255, 8 waves of 32
  const int wv  = tid >> 5;
  const int ln  = tid & 31;

  __shared__ __align__(16) unsigned char stage[8 * 2048]; // 8 waves x 4 x 512B
  __shared__ float redS[8], redQ[8];
  __shared__ float wtab[LL], btab[LL];
  __shared__ float s_mean, s_inv;

  const float* xplane = x + (size_t)bx * HWSZ;

  if (tid < LL) {
    wtab[tid] = weight[tid * CC + c];
    btab[tid] = bias[tid * CC + c];
  }

  // ---------------- Pass 1: async-staged sum / sumsq ----------------
  constexpr int ITERS = HWSZ / (256 * 4);            // 64 chunks of 4KB
  // wave-private LDS base for this lane (low 32 bits of flat shared addr
  // == workgroup-relative LDS byte offset)
  const unsigned ldsbase =
      (unsigned)(uintptr_t)&stage[(wv << 11) + (ln << 4)];
  const unsigned gbase = (unsigned)((wv << 9) + (ln << 4)); // byte off in plane

  float sum = 0.f, sq = 0.f;

  // prime 3 chunks of the 4-deep ring
  ASYNC_LOAD_B128(ldsbase,                 gbase,            xplane);
  ASYNC_LOAD_B128(ldsbase + (1u << 9),     gbase + 4096u,    xplane);
  ASYNC_LOAD_B128(ldsbase + (2u << 9),     gbase + 8192u,    xplane);

  for (int i = 0; i < ITERS; ++i) {
    if (i + 3 < ITERS) {
      const unsigned nlds = ldsbase + ((unsigned)((i + 3) & 3) << 9);
      const unsigned noff = gbase + (unsigned)(i + 3) * 4096u;
      ASYNC_LOAD_B128(nlds, noff, xplane);
      WAIT_ASYNC(3);                                 // chunk i has landed
    } else if (i + 3 == ITERS) {
      WAIT_ASYNC(2);
    } else if (i + 2 == ITERS) {
      WAIT_ASYNC(1);
    } else {
      WAIT_ASYNC(0);
    }
    const vf4 v = *reinterpret_cast<const vf4*>(
        &stage[(wv << 11) + ((i & 3) << 9) + (ln << 4)]);
    sum += v.x + v.y + v.z + v.w;
    sq  += v.x * v.x + v.y * v.y + v.z * v.z + v.w * v.w;
  }

  // wave32 reduction
#pragma unroll
  for (int off = 16; off > 0; off >>= 1) {
    sum += __shfl_down(sum, off, 32);
    sq  += __shfl_down(sq, off, 32);
  }
  if (ln == 0) { redS[wv] = sum; redQ[wv] = sq; }
  __syncthreads();
  if (tid == 0) {
    float S = 0.f, Q = 0.f;
#pragma unroll
    for (int i = 0; i < 8; ++i) { S += redS[i]; Q += redQ[i]; }
    const float mean = S * (1.0f / HWSZ);
    const float var  = Q * (1.0f / HWSZ) - mean * mean;
    s_mean = mean;
    s_inv  = rsqrtf(var + EPSV);
  }
  __syncthreads();
  const float mean = s_mean;
  const float inv  = s_inv;

  // ---------------- Pass 2: normalize + class-conditional affine ----------
  const vf4* xin4 = reinterpret_cast<const vf4*>(xplane);
  vf4* out4 = reinterpret_cast<vf4*>(out) + (size_t)bx * (HWSZ / 4);
  const unsigned* cls4 =
      reinterpret_cast<const unsigned*>(cls + (size_t)n * HWSZ);

#pragma unroll 4
  for (int i = 0; i < ITERS; ++i) {
    const int idx = i * 256 + tid;
    const vf4 v = xin4[idx];
    const unsigned cb = cls4[idx];
    const int k0 = cb & 255, k1 = (cb >> 8) & 255,
              k2 = (cb >> 16) & 255, k3 = cb >> 24;
    vf4 o;
    o.x = (v.x - mean) * inv * wtab[k0] + btab[k0];
    o.y = (v.y - mean) * inv * wtab[k1] + btab[k1];
    o.z = (v.z - mean) * inv * wtab[k2] + btab[k2];
    o.w = (v.w - mean) * inv * wtab[k3] + btab[k3];
    __builtin_nontemporal_store(o, &out4[idx]);
  }
}

extern "C" void kernel_launch(void* const* d_in, const int* in_sizes, int n_in,
                              void* d_out, int out_size, void* d_ws, size_t ws_size,
                              hipStream_t stream) {
  const float* x   = (const float*)d_in[0];   // [N,C,H,W]
  const float* seg = (const float*)d_in[1];   // [N,L,H,W]
  const float* wgt = (const float*)d_in[2];   // [L,C]
  const float* bia = (const float*)d_in[3];   // [L,C]
  float* out = (float*)d_out;                 // [N,C,H,W]
  unsigned char* cls = (unsigned char*)d_ws;  // [N,H,W] uint8 class ids

  const int argmax_threads = NN * (HWSZ / 4);          // 131072
  spade_argmax_kernel<<<(argmax_threads + 255) / 256, 256, 0, stream>>>(seg, cls);
  spade_norm_kernel<<<NN * CC, 256, 0, stream>>>(x, cls, wgt, bia, out);
}